// SECOND_BEAM_SEARCH_50998441673027
// MI455X (gfx1250) — compile-verified
//
#include <hip/hip_runtime.h>
#include <hip/hip_bf16.h>
#include <math.h>

// Problem dims (fixed by the reference)
#define L_  28
#define B_  3
#define H_  8
#define S_  2048
#define D_  128
#define V_  152064
#define KVN_LL 176160768LL              // L*B*H*S*D
#define SLICE_F4 524288                 // H*S*D/4 float4 per (l,b) slice

// Gather tiling
#define GBLK 256                        // threads per block
#define GDEPTH 8                        // async b128 ops per lane per phase
#define GPHASES 8                       // phases per block
#define BLOCKS_PER_SLICE 32             // 32 * 8 * 8 * 256 f4 = 524288 f4 = one slice

__device__ __forceinline__ void ins3(float& v0, float& v1, float& v2,
                                     int& i0, int& i1, int& i2,
                                     float x, int i) {
  if (x > v2) {
    if (x > v0)      { v2=v1; i2=i1; v1=v0; i1=i0; v0=x; i0=i; }
    else if (x > v1) { v2=v1; i2=i1; v1=x;  i1=i; }
    else             { v2=x;  i2=i; }
  }
}

// ---------------------------------------------------------------------------
// Kernel 1: per-row logsumexp + per-row top-3 over V=152064 logits.
// One block per beam row. ~1.8 MB total read — negligible vs the KV gather.
// ---------------------------------------------------------------------------
__global__ void __launch_bounds__(GBLK)
row_stats_kernel(const float* __restrict__ logits,
                 float* __restrict__ wsf,
                 int* __restrict__ wsi) {
  const int row = blockIdx.x;
  const int tid = threadIdx.x;
  const float* x = logits + (size_t)row * V_;

  float m = -INFINITY, s = 0.f;
  float v0 = -INFINITY, v1 = -INFINITY, v2 = -INFINITY;
  int   i0 = -1, i1 = -1, i2 = -1;

  for (int i = tid; i < V_; i += GBLK) {
    float xv = x[i];
    if (xv > m) { s = s * expf(m - xv) + 1.f; m = xv; }
    else        { s += expf(xv - m); }
    ins3(v0, v1, v2, i0, i1, i2, xv, i);
  }

  __shared__ float sm[GBLK], ss[GBLK], sv[GBLK][3];
  __shared__ int   si[GBLK][3];
  sm[tid] = m; ss[tid] = s;
  sv[tid][0] = v0; sv[tid][1] = v1; sv[tid][2] = v2;
  si[tid][0] = i0; si[tid][1] = i1; si[tid][2] = i2;
  __syncthreads();

  for (int stride = GBLK / 2; stride > 0; stride >>= 1) {
    if (tid < stride) {
      // merge (max, sumexp) pairs
      float m2 = sm[tid + stride], s2 = ss[tid + stride];
      float M  = fmaxf(sm[tid], m2);
      ss[tid] = ss[tid] * expf(sm[tid] - M) + s2 * expf(m2 - M);
      sm[tid] = M;
      // merge top-3 triples
      float a0 = sv[tid][0], a1 = sv[tid][1], a2 = sv[tid][2];
      int   x0 = si[tid][0], x1 = si[tid][1], x2 = si[tid][2];
      ins3(a0, a1, a2, x0, x1, x2, sv[tid + stride][0], si[tid + stride][0]);
      ins3(a0, a1, a2, x0, x1, x2, sv[tid + stride][1], si[tid + stride][1]);
      ins3(a0, a1, a2, x0, x1, x2, sv[tid + stride][2], si[tid + stride][2]);
      sv[tid][0] = a0; sv[tid][1] = a1; sv[tid][2] = a2;
      si[tid][0] = x0; si[tid][1] = x1; si[tid][2] = x2;
    }
    __syncthreads();
  }

  if (tid == 0) {
    wsf[row] = sm[0] + logf(ss[0]);        // row logsumexp
    for (int k = 0; k < 3; ++k) {
      wsf[8 + row * 3 + k] = sv[0][k];     // top-3 values
      wsi[row * 3 + k]     = si[0][k];     // top-3 indices
    }
  }
}

// ---------------------------------------------------------------------------
// Kernel 2: global top-3 over the 9 (beam, candidate) scores; writes all
// small outputs and beam_index[3] (parents) to the workspace for the gather.
// ---------------------------------------------------------------------------
__global__ void __launch_bounds__(GBLK)
select_kernel(const float* __restrict__ wsf,
              int* __restrict__ wsi,
              const float* __restrict__ prev,
              const int* __restrict__ save_id,
              float* __restrict__ out) {
  __shared__ int s_parent[3], s_tok[3];

  if (threadIdx.x == 0) {
    float b0 = -INFINITY, b1 = -INFINITY, b2 = -INFINITY;
    int   j0 = -1, j1 = -1, j2 = -1;
    for (int j = 0; j < 9; ++j) {
      int b = j / 3;
      float c = wsf[8 + j] - wsf[b] + prev[b];   // top_k_prob + previous_prob
      ins3(b0, b1, b2, j0, j1, j2, c, j);
    }
    int   js[3] = { j0, j1, j2 };
    float bs[3] = { b0, b1, b2 };
    for (int i = 0; i < 3; ++i) {
      int parent = js[i] / 3;
      int tok    = wsi[js[i]];
      s_parent[i] = parent;
      s_tok[i]    = tok;
      wsi[16 + i] = parent;                       // beam_index for the gather
      out[KVN_LL + i]       = (float)tok;         // top_beam_indices [3,1]
      out[KVN_LL + 198 + i] = bs[i];              // top_beam_prob   [3,1]
    }
    out[KVN_LL + 201] = (float)s_tok[0];          // max_logits_idx
  }
  __syncthreads();

  // new_save_id [3, 65] = concat(save_id[parent], token)
  for (int e = threadIdx.x; e < 3 * 65; e += blockDim.x) {
    int r = e / 65, c = e % 65;
    int val = (c < 64) ? save_id[s_parent[r] * 64 + c] : s_tok[r];
    out[KVN_LL + 3 + e] = (float)val;
  }
}

// ---------------------------------------------------------------------------
// Kernel 3: the 704.6 MB KV beam gather via CDNA5 async global<->LDS DMA,
// double-buffered so each wave keeps 16 async b128 ops (8 KB, mixed
// read+write streams) in flight between ASYNCcnt drains.
//
// Pipeline per wave (full drains only — ASYNCcnt completes out-of-order
// between loads and stores, so partial waits are not reliable):
//   preload buf0 ; wait 0
//   for p: issue stores(buf[p&1]) + issue loads(buf[(p+1)&1]) ; wait 0
// Buffer reuse is safe: any buffer being refilled was drained at the
// previous wait 0. Each lane owns its LDS slots: no barriers needed.
// ---------------------------------------------------------------------------
__global__ void __launch_bounds__(GBLK)
kv_gather_kernel(const float* __restrict__ kv,
                 const int* __restrict__ wsi,
                 float* __restrict__ out) {
  __shared__ alignas(16) char stage[2][GBLK * 16 * GDEPTH];   // 2 x 32 KB

  const int tid   = threadIdx.x;
  const int bid   = blockIdx.x;
  const int slice = bid / BLOCKS_PER_SLICE;   // 0..83  -> (layer, out_beam)
  const int sub   = bid % BLOCKS_PER_SLICE;
  const int l     = slice / B_;
  const int ob    = slice % B_;
  const int sb    = wsi[16 + ob];             // surviving parent beam

  const size_t chunk_f4 = SLICE_F4 / BLOCKS_PER_SLICE;   // 16384 float4
  const float4* __restrict__ src =
      (const float4*)kv + (size_t)(l * B_ + sb) * SLICE_F4 + (size_t)sub * chunk_f4;
  float4* __restrict__ dst =
      (float4*)out + (size_t)(l * B_ + ob) * SLICE_F4 + (size_t)sub * chunk_f4;

  const unsigned lds0 = (unsigned)(size_t)(&stage[0][0]);  // low 32b = LDS offset
  const unsigned lds1 = (unsigned)(size_t)(&stage[1][0]);

  // ---- prologue: fill buffer 0 with phase 0 ----
#pragma unroll
  for (int d = 0; d < GDEPTH; ++d) {
    const float4* g = src + d * GBLK + tid;
    unsigned la = lds0 + (unsigned)((d * GBLK + tid) * 16);
    asm volatile("global_load_async_to_lds_b128 %0, %1, off"
                 :: "v"(la), "v"(g) : "memory");
  }
  asm volatile("s_wait_asynccnt 0" ::: "memory");

  // ---- steady state: store phase p while loading phase p+1 ----
  for (int p = 0; p < GPHASES; ++p) {
    const unsigned cur = (p & 1) ? lds1 : lds0;
    const int base = p * (GDEPTH * GBLK);
#pragma unroll
    for (int d = 0; d < GDEPTH; ++d) {
      float4* g = dst + base + d * GBLK + tid;
      unsigned la = cur + (unsigned)((d * GBLK + tid) * 16);
      asm volatile("global_store_async_from_lds_b128 %0, %1, off"
                   :: "v"(g), "v"(la) : "memory");
    }
    if (p + 1 < GPHASES) {
      const unsigned nxt = (p & 1) ? lds0 : lds1;
      const int nbase = (p + 1) * (GDEPTH * GBLK);
#pragma unroll
      for (int d = 0; d < GDEPTH; ++d) {
        const float4* g = src + nbase + d * GBLK + tid;
        unsigned la = nxt + (unsigned)((d * GBLK + tid) * 16);
        asm volatile("global_load_async_to_lds_b128 %0, %1, off"
                     :: "v"(la), "v"(g) : "memory");
      }
    }
    asm volatile("s_wait_asynccnt 0" ::: "memory");
  }
}

// ---------------------------------------------------------------------------
extern "C" void kernel_launch(void* const* d_in, const int* in_sizes, int n_in,
                              void* d_out, int out_size, void* d_ws, size_t ws_size,
                              hipStream_t stream) {
  const float* kv      = (const float*)d_in[0];   // [L,B,H,S,D] fp32
  const float* logits  = (const float*)d_in[1];   // [B,V] fp32
  const int*   save_id = (const int*)d_in[2];     // [B,64] int32
  const float* prev    = (const float*)d_in[3];   // [B,1] fp32
  float* out = (float*)d_out;

  float* wsf = (float*)d_ws;                      // [0..2]=lse, [8..16]=top3 vals
  int*   wsi = (int*)d_ws + 64;                   // [0..8]=top3 idx, [16..18]=beam_index

  row_stats_kernel<<<B_, GBLK, 0, stream>>>(logits, wsf, wsi);
  select_kernel<<<1, GBLK, 0, stream>>>(wsf, wsi, prev, save_id, out);
  kv_gather_kernel<<<L_ * B_ * BLOCKS_PER_SLICE, GBLK, 0, stream>>>(kv, wsi, out);
}